// EVPointNetEncoder_51556787421344
// MI455X (gfx1250) — compile-verified
//
#include <hip/hip_runtime.h>

#define KNN 20
#define NEG_SLOPE 0.2f
#define EPSV 1e-6f
#define KBLK 128

typedef float v2f __attribute__((ext_vector_type(2)));
typedef float v8f __attribute__((ext_vector_type(8)));

// ---------------------------------------------------------------------------
// KNN: one thread per point, candidates tiled through LDS, top-20 insertion.
// ---------------------------------------------------------------------------
__global__ void knn_kernel(const float* __restrict__ pts, int* __restrict__ idx, int N) {
  __shared__ float sx[256], sy[256], sz[256];
  int b = blockIdx.y;
  int n = blockIdx.x * blockDim.x + threadIdx.x;
  const float* px = pts + (size_t)b * 3 * N;
  float x0 = px[n], y0 = px[N + n], z0 = px[2 * N + n];
  float best[KNN];
  int bidx[KNN];
#pragma unroll
  for (int i = 0; i < KNN; ++i) { best[i] = -3.4e38f; bidx[i] = 0; }
  for (int m0 = 0; m0 < N; m0 += 256) {
    __syncthreads();
    sx[threadIdx.x] = px[m0 + threadIdx.x];
    sy[threadIdx.x] = px[N + m0 + threadIdx.x];
    sz[threadIdx.x] = px[2 * N + m0 + threadIdx.x];
    __syncthreads();
    for (int j = 0; j < 256; ++j) {
      float dx = sx[j] - x0, dy = sy[j] - y0, dz = sz[j] - z0;
      float nd = -(dx * dx + dy * dy + dz * dz);
      if (nd > best[KNN - 1]) {
        int pos = KNN - 1;
        while (pos > 0 && best[pos - 1] < nd) {
          best[pos] = best[pos - 1];
          bidx[pos] = bidx[pos - 1];
          --pos;
        }
        best[pos] = nd;
        bidx[pos] = m0 + j;
      }
    }
  }
  int* ob = idx + ((size_t)b * N + n) * KNN;
#pragma unroll
  for (int i = 0; i < KNN; ++i) ob[i] = bidx[i];
}

// ---------------------------------------------------------------------------
// Fused edge-feature + (W_pos,U_pos) VN-LeakyReLU + mean over k.
// One wave per point, lane = output channel (21 of 32 lanes active).
// ---------------------------------------------------------------------------
__global__ void edge_pos_kernel(const float* __restrict__ pts, const int* __restrict__ idx,
                                const float* __restrict__ Wp, const float* __restrict__ Up,
                                const float* __restrict__ scale, float* __restrict__ h1, int N) {
  int wave = threadIdx.x >> 5, lane = threadIdx.x & 31;
  int pidx = blockIdx.x * 4 + wave;  // grid covers exactly B*N/4 blocks
  int b = pidx / N, n = pidx % N;
  const float* px = pts + (size_t)b * 3 * N;
  float s = scale[0];
  float cx = px[n], cy = px[N + n], cz = px[2 * N + n];
  int cw = (lane < 21) ? lane : 20;  // clamp, select later
  float w0 = Wp[cw * 3 + 0], w1 = Wp[cw * 3 + 1], w2 = Wp[cw * 3 + 2];
  float u0 = Up[cw * 3 + 0], u1 = Up[cw * 3 + 1], u2 = Up[cw * 3 + 2];
  float a0 = 0, a1 = 0, a2 = 0;
  const int* ib = idx + ((size_t)b * N + n) * KNN;
  for (int k = 0; k < KNN; ++k) {
    int m = ib[k];
    float nx = px[m], ny = px[N + m], nz = px[2 * N + m];
    // channels: c0 = (nbr-ctr)*s, c1 = ctr*s, c2 = cross(nbr,ctr)*s
    float ex = (nx - cx) * s, ey = (ny - cy) * s, ez = (nz - cz) * s;
    float tx = cx * s, ty = cy * s, tz = cz * s;
    float rx = (ny * cz - nz * cy) * s;
    float ry = (nz * cx - nx * cz) * s;
    float rz = (nx * cy - ny * cx) * s;
    float p0 = w0 * ex + w1 * tx + w2 * rx;
    float p1 = w0 * ey + w1 * ty + w2 * ry;
    float p2 = w0 * ez + w1 * tz + w2 * rz;
    float d0 = u0 * ex + u1 * tx + u2 * rx;
    float d1 = u0 * ey + u1 * ty + u2 * ry;
    float d2 = u0 * ez + u1 * tz + u2 * rz;
    float dot = p0 * d0 + p1 * d1 + p2 * d2;
    float dn = d0 * d0 + d1 * d1 + d2 * d2 + EPSV;
    float t = dot / dn;
    float q0 = (dot >= 0.f) ? p0 : (p0 - t * d0);
    float q1 = (dot >= 0.f) ? p1 : (p1 - t * d1);
    float q2 = (dot >= 0.f) ? p2 : (p2 - t * d2);
    a0 += NEG_SLOPE * p0 + (1.f - NEG_SLOPE) * q0;
    a1 += NEG_SLOPE * p1 + (1.f - NEG_SLOPE) * q1;
    a2 += NEG_SLOPE * p2 + (1.f - NEG_SLOPE) * q2;
  }
  if (lane < 21) {
    float inv = 1.f / (float)KNN;
    size_t base = ((size_t)b * 21 + lane) * 3 * (size_t)N + n;
    h1[base] = a0 * inv;
    h1[base + N] = a1 * inv;
    h1[base + 2 * (size_t)N] = a2 * inv;
  }
}

// ---------------------------------------------------------------------------
// Dual GEMM: P[b] = W @ X[b], D[b] = U @ X[b]  (Cout x NC) via
// V_WMMA_F32_16X16X4_F32. 256 threads = 8 waves; block owns a 32-row M-strip
// (two 16-row tiles), staged in LDS per KBLK slice; each wave owns one 16-col
// N-tile and shares each B load pair across 4 WMMAs (2 M-tiles x {W,U}).
// Row padding (+2) de-conflicts LDS banks while keeping 8B alignment.
// LDS fill uses clamped (always-in-bounds) loads + cndmask zeroing so the
// fill loop has no EXEC divergence.
// ---------------------------------------------------------------------------
template <bool HAS_U>
__global__ __launch_bounds__(256) void gemm_wmma_k(const float* __restrict__ Wm,
                                                   const float* __restrict__ Um,
                                                   const float* __restrict__ X,
                                                   float* __restrict__ P,
                                                   float* __restrict__ D, int Cout, int K,
                                                   int NC, size_t ldxb, size_t ldyb) {
  __shared__ float sW[32][KBLK + 2];
  __shared__ float sU[HAS_U ? 32 : 1][KBLK + 2];

  int lane = threadIdx.x & 31;
  int wave = threadIdx.x >> 5;
  int ntile = blockIdx.x * 8 + wave;
  int mbase = blockIdx.y * 32;
  int b = blockIdx.z;

  const float* Xb = X + (size_t)b * ldxb;
  int n = ntile * 16 + (lane & 15);
  int khalf = (lane >> 4) * 2;  // lanes 0-15: K {0,1}; lanes 16-31: K {2,3}
  int mrow = lane & 15;

  v8f accP0 = {}, accP1 = {}, accD0 = {}, accD1 = {};
  int Kpad = (K + 3) & ~3;

  for (int kb = 0; kb < Kpad; kb += KBLK) {
    __syncthreads();
    for (int i = threadIdx.x; i < 32 * KBLK; i += 256) {
      int m = i / KBLK, kk = i % KBLK;
      int gm = mbase + m, k = kb + kk;
      bool ok = (gm < Cout) && (k < K);
      int gmc = (gm < Cout) ? gm : (Cout - 1);  // clamped: load always in-bounds
      int kc = (k < K) ? k : (K - 1);
      size_t a = (size_t)gmc * K + kc;
      float w = Wm[a];
      sW[m][kk] = ok ? w : 0.0f;
      if (HAS_U) {
        float u = Um[a];
        sU[m][kk] = ok ? u : 0.0f;
      }
    }
    __syncthreads();

    int rem = K - kb;  // valid K remaining (may exceed KBLK)
    int klim = Kpad - kb;
    if (klim > KBLK) klim = KBLK;
    int kfull = (rem < KBLK) ? (rem & ~3) : KBLK;  // wave-uniform trip count

    const float* bp = Xb + (size_t)(kb + khalf) * NC + n;
    int k0 = 0;
    for (; k0 < kfull; k0 += 4) {  // unguarded hot loop
      v2f bb, a0, a1;
      bb.x = bp[0];
      bb.y = bp[NC];
      bp += (size_t)4 * NC;
      a0.x = sW[mrow][k0 + khalf];
      a0.y = sW[mrow][k0 + khalf + 1];
      a1.x = sW[16 + mrow][k0 + khalf];
      a1.y = sW[16 + mrow][k0 + khalf + 1];
      accP0 = __builtin_amdgcn_wmma_f32_16x16x4_f32(false, a0, false, bb, (short)0, accP0,
                                                    false, false);
      accP1 = __builtin_amdgcn_wmma_f32_16x16x4_f32(false, a1, false, bb, (short)0, accP1,
                                                    false, false);
      if (HAS_U) {
        v2f u0, u1;
        u0.x = sU[mrow][k0 + khalf];
        u0.y = sU[mrow][k0 + khalf + 1];
        u1.x = sU[16 + mrow][k0 + khalf];
        u1.y = sU[16 + mrow][k0 + khalf + 1];
        accD0 = __builtin_amdgcn_wmma_f32_16x16x4_f32(false, u0, false, bb, (short)0, accD0,
                                                      false, false);
        accD1 = __builtin_amdgcn_wmma_f32_16x16x4_f32(false, u1, false, bb, (short)0, accD1,
                                                      false, false);
      }
    }
    if (k0 < klim) {  // tail: 1-3 valid k, once per layer at most
      int kg = kb + k0 + khalf;
      int kg0 = (kg < K) ? kg : (K - 1);
      int kg1 = (kg + 1 < K) ? (kg + 1) : (K - 1);
      v2f bb, a0, a1;
      float bx = Xb[(size_t)kg0 * NC + n];
      float by = Xb[(size_t)kg1 * NC + n];
      bb.x = (kg < K) ? bx : 0.0f;
      bb.y = (kg + 1 < K) ? by : 0.0f;
      a0.x = sW[mrow][k0 + khalf];
      a0.y = sW[mrow][k0 + khalf + 1];
      a1.x = sW[16 + mrow][k0 + khalf];
      a1.y = sW[16 + mrow][k0 + khalf + 1];
      accP0 = __builtin_amdgcn_wmma_f32_16x16x4_f32(false, a0, false, bb, (short)0, accP0,
                                                    false, false);
      accP1 = __builtin_amdgcn_wmma_f32_16x16x4_f32(false, a1, false, bb, (short)0, accP1,
                                                    false, false);
      if (HAS_U) {
        v2f u0, u1;
        u0.x = sU[mrow][k0 + khalf];
        u0.y = sU[mrow][k0 + khalf + 1];
        u1.x = sU[16 + mrow][k0 + khalf];
        u1.y = sU[16 + mrow][k0 + khalf + 1];
        accD0 = __builtin_amdgcn_wmma_f32_16x16x4_f32(false, u0, false, bb, (short)0, accD0,
                                                      false, false);
        accD1 = __builtin_amdgcn_wmma_f32_16x16x4_f32(false, u1, false, bb, (short)0, accD1,
                                                      false, false);
      }
    }
  }

  // D-layout store: vgpr r -> M = r + 8*(lane/16), N = lane&15
  size_t ybase = (size_t)b * ldyb + (size_t)ntile * 16 + (lane & 15);
  int msub = (lane >> 4) * 8;
#pragma unroll
  for (int r = 0; r < 8; ++r) {
    int m0 = mbase + msub + r;
    int m1 = m0 + 16;
    if (m0 < Cout) P[ybase + (size_t)m0 * NC] = accP0[r];
    if (m1 < Cout) P[ybase + (size_t)m1 * NC] = accP1[r];
    if (HAS_U) {
      if (m0 < Cout) D[ybase + (size_t)m0 * NC] = accD0[r];
      if (m1 < Cout) D[ybase + (size_t)m1 * NC] = accD1[r];
    }
  }
}

// ---------------------------------------------------------------------------
// Pointwise vector-neuron LeakyReLU on P,D laid out [B,C,3,N] (compact).
// ---------------------------------------------------------------------------
__global__ void vn_lrelu_kernel(const float* __restrict__ P, const float* __restrict__ D,
                                float* __restrict__ O, int C, int N, int total) {
  int t = blockIdx.x * blockDim.x + threadIdx.x;
  if (t >= total) return;
  int n = t % N;
  int r = t / N;
  int c = r % C;
  int b = r / C;
  size_t base = (((size_t)b * C + c) * 3) * (size_t)N + n;
  float p0 = P[base], p1 = P[base + N], p2 = P[base + 2 * (size_t)N];
  float d0 = D[base], d1 = D[base + N], d2 = D[base + 2 * (size_t)N];
  float dot = p0 * d0 + p1 * d1 + p2 * d2;
  float dn = d0 * d0 + d1 * d1 + d2 * d2 + EPSV;
  float s = dot / dn;
  float q0 = (dot >= 0.f) ? p0 : (p0 - s * d0);
  float q1 = (dot >= 0.f) ? p1 : (p1 - s * d1);
  float q2 = (dot >= 0.f) ? p2 : (p2 - s * d2);
  O[base] = NEG_SLOPE * p0 + (1.f - NEG_SLOPE) * q0;
  O[base + N] = NEG_SLOPE * p1 + (1.f - NEG_SLOPE) * q1;
  O[base + 2 * (size_t)N] = NEG_SLOPE * p2 + (1.f - NEG_SLOPE) * q2;
}

// ---------------------------------------------------------------------------
// Row reductions (mean / max) over contiguous N.
// ---------------------------------------------------------------------------
__global__ void rowmean_kernel(const float* __restrict__ src, float* __restrict__ dst,
                               int C3, int N, size_t bstride) {
  __shared__ float red[256];
  int row = blockIdx.x;  // over B*C3
  int b = row / C3, r = row % C3;
  const float* p = src + (size_t)b * bstride + (size_t)r * N;
  float s = 0.f;
  for (int i = threadIdx.x; i < N; i += 256) s += p[i];
  red[threadIdx.x] = s;
  __syncthreads();
  for (int off = 128; off > 0; off >>= 1) {
    if (threadIdx.x < off) red[threadIdx.x] += red[threadIdx.x + off];
    __syncthreads();
  }
  if (threadIdx.x == 0) dst[row] = red[0] / (float)N;
}

__global__ void rowmax_kernel(const float* __restrict__ src, float* __restrict__ dst, int N) {
  __shared__ float red[256];
  int row = blockIdx.x;
  const float* p = src + (size_t)row * N;
  float s = -3.4e38f;
  for (int i = threadIdx.x; i < N; i += 256) s = fmaxf(s, p[i]);
  red[threadIdx.x] = s;
  __syncthreads();
  for (int off = 128; off > 0; off >>= 1) {
    if (threadIdx.x < off) red[threadIdx.x] = fmaxf(red[threadIdx.x], red[threadIdx.x + off]);
    __syncthreads();
  }
  if (threadIdx.x == 0) dst[row] = red[0];
}

// ---------------------------------------------------------------------------
// Small EV layers on [B,Cin,3] (v contiguous). One thread per (b, o).
// ---------------------------------------------------------------------------
__global__ void ev_small(const float* __restrict__ Wm, const float* __restrict__ Um,
                         const float* __restrict__ X, float* __restrict__ O, int Cin,
                         int Cout, int total) {
  int t = blockIdx.x * blockDim.x + threadIdx.x;
  if (t >= total) return;
  int o = t % Cout, b = t / Cout;
  const float* xb = X + (size_t)b * Cin * 3;
  float p0 = 0, p1 = 0, p2 = 0, d0 = 0, d1 = 0, d2 = 0;
  for (int c = 0; c < Cin; ++c) {
    float x0 = xb[c * 3], x1 = xb[c * 3 + 1], x2 = xb[c * 3 + 2];
    float w = Wm[(size_t)o * Cin + c];
    p0 += w * x0; p1 += w * x1; p2 += w * x2;
    if (Um) {
      float u = Um[(size_t)o * Cin + c];
      d0 += u * x0; d1 += u * x1; d2 += u * x2;
    }
  }
  float r0 = p0, r1 = p1, r2 = p2;
  if (Um) {
    float dot = p0 * d0 + p1 * d1 + p2 * d2;
    float dn = d0 * d0 + d1 * d1 + d2 * d2 + EPSV;
    float s = dot / dn;
    float q0 = (dot >= 0.f) ? p0 : (p0 - s * d0);
    float q1 = (dot >= 0.f) ? p1 : (p1 - s * d1);
    float q2 = (dot >= 0.f) ? p2 : (p2 - s * d2);
    r0 = NEG_SLOPE * p0 + (1.f - NEG_SLOPE) * q0;
    r1 = NEG_SLOPE * p1 + (1.f - NEG_SLOPE) * q1;
    r2 = NEG_SLOPE * p2 + (1.f - NEG_SLOPE) * q2;
  }
  float* ob = O + ((size_t)b * Cout + o) * 3;
  ob[0] = r0; ob[1] = r1; ob[2] = r2;
}

// h2 = concat([hc1 (21ch), broadcast gsmall (21ch)], ch) -> [B,42,3,N]
__global__ void h2fill_kernel(const float* __restrict__ hc1, const float* __restrict__ gs,
                              float* __restrict__ h2, int N, int total) {
  int t = blockIdx.x * blockDim.x + threadIdx.x;
  if (t >= total) return;
  int n = t % N;
  int r = t / N;
  int v = r % 3; r /= 3;
  int c = r % 42;
  int b = r / 42;
  float val = (c < 21) ? hc1[((((size_t)b * 21 + c) * 3) + v) * (size_t)N + n]
                       : gs[((size_t)b * 21 + (c - 21)) * 3 + v];
  h2[((((size_t)b * 42 + c) * 3) + v) * (size_t)N + n] = val;
}

// fill H5 channels 341..681 with broadcast of hm [B,341,3]
__global__ void h5bcast_kernel(const float* __restrict__ hm, float* __restrict__ h5, int N,
                               int total) {
  int t = blockIdx.x * blockDim.x + threadIdx.x;
  if (t >= total) return;
  int n = t % N;
  int r = t / N;
  int v = r % 3; r /= 3;
  int c = r % 341;
  int b = r / 341;
  h5[((((size_t)b * 682 + 341 + c) * 3) + v) * (size_t)N + n] =
      hm[((size_t)b * 341 + c) * 3 + v];
}

// inv_feat[b,i,kv,n] = sum_j h5[b,i,j,n] * z0[b,j,kv,n]
__global__ void invfeat_kernel(const float* __restrict__ h5, const float* __restrict__ z0,
                               float* __restrict__ inv, int N, int total) {
  int t = blockIdx.x * blockDim.x + threadIdx.x;
  if (t >= total) return;
  int n = t % N;
  int r = t / N;
  int i = r % 682;
  int b = r / 682;
  const float* hb = h5 + (((size_t)b * 682 + i) * 3) * (size_t)N + n;
  float h0 = hb[0], h1v = hb[(size_t)N], h2v = hb[2 * (size_t)N];
  const float* zb = z0 + (size_t)b * 9 * N + n;
  float* ob = inv + (((size_t)b * 682 + i) * 3) * (size_t)N + n;
#pragma unroll
  for (int kv = 0; kv < 3; ++kv) {
    float v = h0 * zb[(size_t)(0 * 3 + kv) * N] + h1v * zb[(size_t)(1 * 3 + kv) * N] +
              h2v * zb[(size_t)(2 * 3 + kv) * N];
    ob[(size_t)kv * N] = v;
  }
}

extern "C" void kernel_launch(void* const* d_in, const int* in_sizes, int n_in, void* d_out,
                              int out_size, void* d_ws, size_t ws_size, hipStream_t stream) {
  const int B = 8, N = 4096, NC = 3 * N;
  const float* x = (const float*)d_in[0];
  const float* scale = (const float*)d_in[1];
  const float* W_pos = (const float*)d_in[2];
  const float* U_pos = (const float*)d_in[3];
  const float* W_c1 = (const float*)d_in[4];
  const float* U_c1 = (const float*)d_in[5];
  const float* W_f1 = (const float*)d_in[6];
  const float* U_f1 = (const float*)d_in[7];
  const float* W_f2 = (const float*)d_in[8];
  const float* U_f2 = (const float*)d_in[9];
  const float* W_f3 = (const float*)d_in[10];
  const float* U_f3 = (const float*)d_in[11];
  const float* W_fc1 = (const float*)d_in[12];
  const float* U_fc1 = (const float*)d_in[13];
  const float* W_fc2 = (const float*)d_in[14];
  const float* U_fc2 = (const float*)d_in[15];
  const float* W_fc3 = (const float*)d_in[16];
  const float* W_c2 = (const float*)d_in[17];
  const float* U_c2 = (const float*)d_in[18];
  const float* W_c3 = (const float*)d_in[19];
  const float* W_s1 = (const float*)d_in[20];
  const float* U_s1 = (const float*)d_in[21];
  const float* W_s2 = (const float*)d_in[22];
  const float* U_s2 = (const float*)d_in[23];
  const float* W_sl = (const float*)d_in[24];
  (void)in_sizes; (void)n_in; (void)out_size; (void)ws_size;

  float* out = (float*)d_out;          // [B, 2046]
  float* z0 = out + (size_t)B * 2046;  // [B, 3, 3, N]
  float* inv = z0 + (size_t)B * 9 * N; // [B, 682, 3, N]

  // workspace layout (floats)
  float* W = (float*)d_ws;
  int* IDX = (int*)d_ws;
  size_t off = (size_t)B * N * KNN;                       // idx ints
  float* H1 = W + off;  off += (size_t)B * 21 * 3 * N;    // [B,21,3,N]
  float* HC1 = W + off; off += (size_t)B * 21 * 3 * N;    // [B,21,3,N]
  float* H2 = W + off;  off += (size_t)B * 42 * 3 * N;    // [B,42,3,N]
  float* T3 = W + off;  off += (size_t)B * 42 * 3 * N;    // [B,42,3,N]
  float* H5 = W + off;  off += (size_t)B * 682 * 3 * N;   // [B,682,3,N]
  float* T1 = W + off;  off += (size_t)B * 341 * 3 * N;   // P scratch
  float* T2 = W + off;  off += (size_t)B * 341 * 3 * N;   // D scratch
  float* G3 = W + off;  off += (size_t)B * 341 * 3 * N;   // [B,341,3,N]
  float* SM = W + off;  off += 32768;                     // small tensors
  float* gmean = SM;          // 8*341*3
  float* fa = SM + 8192;      // 8*170*3
  float* fb = SM + 12800;     // 8*85*3
  float* gs = SM + 15360;     // 8*21*3
  float* hm = SM + 16384;     // 8*341*3

  auto gemm2 = [&](const float* Wm, const float* Um, const float* X, float* P, float* D,
                   int Cout, int K, size_t ldxb, size_t ldyb) {
    dim3 grid(NC / 16 / 8, (Cout + 31) / 32, B);
    gemm_wmma_k<true><<<grid, 256, 0, stream>>>(Wm, Um, X, P, D, Cout, K, NC, ldxb, ldyb);
  };
  auto gemm1 = [&](const float* Wm, const float* X, float* P, int Cout, int K, size_t ldxb,
                   size_t ldyb) {
    dim3 grid(NC / 16 / 8, (Cout + 31) / 32, B);
    gemm_wmma_k<false><<<grid, 256, 0, stream>>>(Wm, nullptr, X, P, nullptr, Cout, K, NC,
                                                 ldxb, ldyb);
  };
  auto lrelu = [&](const float* P, const float* D, float* O, int C) {
    int total = B * C * N;
    vn_lrelu_kernel<<<(total + 255) / 256, 256, 0, stream>>>(P, D, O, C, N, total);
  };

  // 1) KNN + fused edge/pos layer
  knn_kernel<<<dim3(N / 256, B), 256, 0, stream>>>(x, IDX, N);
  edge_pos_kernel<<<(B * N) / 4, 128, 0, stream>>>(x, IDX, W_pos, U_pos, scale, H1, N);

  // 2) c1: 21 -> 21
  gemm2(W_c1, U_c1, H1, T1, T2, 21, 21, (size_t)21 * NC, (size_t)21 * NC);
  lrelu(T1, T2, HC1, 21);

  // 3) EVSTN feature branch: f1 (21->21), f2 (21->42), f3 (42->341)
  gemm2(W_f1, U_f1, HC1, T1, T2, 21, 21, (size_t)21 * NC, (size_t)21 * NC);
  lrelu(T1, T2, T3, 21);
  gemm2(W_f2, U_f2, T3, T1, T2, 42, 21, (size_t)21 * NC, (size_t)42 * NC);
  lrelu(T1, T2, H2, 42);  // H2 used as scratch here (overwritten later)
  gemm2(W_f3, U_f3, H2, T1, T2, 341, 42, (size_t)42 * NC, (size_t)341 * NC);
  lrelu(T1, T2, G3, 341);
  rowmean_kernel<<<B * 341 * 3, 256, 0, stream>>>(G3, gmean, 341 * 3, N, (size_t)341 * NC);
  ev_small<<<(B * 170 + 127) / 128, 128, 0, stream>>>(W_fc1, U_fc1, gmean, fa, 341, 170, B * 170);
  ev_small<<<(B * 85 + 127) / 128, 128, 0, stream>>>(W_fc2, U_fc2, fa, fb, 170, 85, B * 85);
  ev_small<<<(B * 21 + 127) / 128, 128, 0, stream>>>(W_fc3, nullptr, fb, gs, 85, 21, B * 21);

  // 4) concat [h, g] -> 42ch, c2 (42->42), c3 (42->341, linear, into H5 lower half)
  {
    int total = B * 42 * 3 * N;
    h2fill_kernel<<<(total + 255) / 256, 256, 0, stream>>>(HC1, gs, H2, N, total);
  }
  gemm2(W_c2, U_c2, H2, T1, T2, 42, 42, (size_t)42 * NC, (size_t)42 * NC);
  lrelu(T1, T2, T3, 42);
  gemm1(W_c3, T3, H5, 341, 42, (size_t)42 * NC, (size_t)682 * NC);

  // 5) global-mean concat -> 682ch
  rowmean_kernel<<<B * 341 * 3, 256, 0, stream>>>(H5, hm, 341 * 3, N, (size_t)682 * NC);
  {
    int total = B * 341 * 3 * N;
    h5bcast_kernel<<<(total + 255) / 256, 256, 0, stream>>>(hm, H5, N, total);
  }

  // 6) std-feature head: s1 (682->341), s2 (341->170), sl (170->3)
  gemm2(W_s1, U_s1, H5, T1, T2, 341, 682, (size_t)682 * NC, (size_t)341 * NC);
  lrelu(T1, T2, G3, 341);
  gemm2(W_s2, U_s2, G3, T1, T2, 170, 341, (size_t)341 * NC, (size_t)170 * NC);
  lrelu(T1, T2, T1, 170);  // elementwise-per-point, in-place safe
  gemm1(W_sl, T1, z0, 3, 170, (size_t)170 * NC, (size_t)9 * N);

  // 7) invariant contraction + global max
  {
    int total = B * 682 * N;
    invfeat_kernel<<<(total + 255) / 256, 256, 0, stream>>>(H5, z0, inv, N, total);
  }
  rowmax_kernel<<<B * 2046, 256, 0, stream>>>(inv, out, N);
}